// GNNModel_46772193853691
// MI455X (gfx1250) — compile-verified
//
#include <hip/hip_runtime.h>

typedef float v2f __attribute__((ext_vector_type(2)));
typedef float v8f __attribute__((ext_vector_type(8)));

#define N_NODES  50000
#define N_EDGES  800000
#define IN_DIM   128
#define HID      64
#define OUT_DIM  2
#define N_GRAPHS 64

// ---------------------------------------------------------------------------
// Degree handling: deg starts at 1.0 (self loop), +1 per incoming edge,
// then dinv = rsqrt(deg) in place.
// ---------------------------------------------------------------------------
__global__ void deg_init_kernel(float* deg, int n) {
    int i = blockIdx.x * blockDim.x + threadIdx.x;
    if (i < n) deg[i] = 1.0f;
}

__global__ void deg_edges_kernel(const long long* __restrict__ col, float* deg, int E) {
    int i = blockIdx.x * blockDim.x + threadIdx.x;
    if (i < E) unsafeAtomicAdd(&deg[(int)col[i]], 1.0f);
}

__global__ void rsqrt_inplace_kernel(float* d, int n) {
    int i = blockIdx.x * blockDim.x + threadIdx.x;
    if (i < n) d[i] = rsqrtf(d[i]);
}

// ---------------------------------------------------------------------------
// FP32 WMMA GEMM: C[M,N] = A[M,K] @ B[K,N].
// One 16x16 output tile per wave via V_WMMA_F32_16X16X4_F32.
// Block = 128 threads (4 waves) -> one 16-row strip, N = 64 = 4 tiles.
// Requires M % 16 == 0, N == 64, K % 4 == 0. EXEC is all ones everywhere.
//
// A fragment (16x4 f32): lane L holds M = L%16; v[0]=K(k0+2*half), v[1]=+1.
// B fragment (4x16 f32): lane L holds N = L%16; halves split K the same way.
// D (16x16 f32): VGPR r -> M = r + 8*half, N = L%16.
// ---------------------------------------------------------------------------
__global__ __launch_bounds__(128) void gemm_wmma_f32(
    const float* __restrict__ A, const float* __restrict__ B,
    float* __restrict__ C, int M, int N, int K)
{
    const int lane  = threadIdx.x & 31;
    const int wave  = threadIdx.x >> 5;
    const int l16   = lane & 15;
    const int half  = lane >> 4;           // 0: lanes 0-15, 1: lanes 16-31
    const int mBase = blockIdx.x * 16;
    const int nBase = wave * 16;

    const float* arow = A + (size_t)(mBase + l16) * K;

    v8f acc = {0.f, 0.f, 0.f, 0.f, 0.f, 0.f, 0.f, 0.f};
    for (int k0 = 0; k0 < K; k0 += 4) {
        const int ka = k0 + 2 * half;
        v2f a;
        a.x = arow[ka];
        a.y = arow[ka + 1];
        v2f b;
        b.x = B[(size_t)ka * N + nBase + l16];
        b.y = B[(size_t)(ka + 1) * N + nBase + l16];
        acc = __builtin_amdgcn_wmma_f32_16x16x4_f32(
            /*neg_a=*/false, a, /*neg_b=*/false, b,
            /*c_mod=*/(short)0, acc, /*reuse_a=*/false, /*reuse_b=*/false);
    }

#pragma unroll
    for (int r = 0; r < 8; ++r) {
        const int row = mBase + r + 8 * half;
        C[(size_t)row * N + nBase + l16] = acc[r];
    }
}

// ---------------------------------------------------------------------------
// Edge scatter: agg[col[e]] += h[row[e]] * dinv[row]*dinv[col].
// One thread per (edge, 4-feature group): float4 gather + 4 hw float atomics.
// ---------------------------------------------------------------------------
__global__ __launch_bounds__(256) void edge_scatter_kernel(
    const long long* __restrict__ rowi, const long long* __restrict__ coli,
    const float* __restrict__ dinv, const float* __restrict__ h,
    float* agg, int E)
{
    int idx = blockIdx.x * blockDim.x + threadIdx.x;
    if (idx >= E * (HID / 4)) return;
    const int e = idx >> 4;            // HID/4 == 16 groups per edge
    const int q = (idx & 15) << 2;     // feature offset 0..60
    const int s = (int)rowi[e];
    const int t = (int)coli[e];
    const float norm = dinv[s] * dinv[t];
    const float4 v = *(const float4*)(h + (size_t)s * HID + q);
    float* dst = agg + (size_t)t * HID + q;
    unsafeAtomicAdd(dst + 0, v.x * norm);
    unsafeAtomicAdd(dst + 1, v.y * norm);
    unsafeAtomicAdd(dst + 2, v.z * norm);
    unsafeAtomicAdd(dst + 3, v.w * norm);
}

// ---------------------------------------------------------------------------
// Epilogue: agg = relu(agg + hlin*dinv^2 + bias)  (self-loop folded in).
// ---------------------------------------------------------------------------
__global__ void epilogue_kernel(float* agg, const float* __restrict__ hlin,
                                const float* __restrict__ dinv,
                                const float* __restrict__ bias, int total)
{
    int i = blockIdx.x * blockDim.x + threadIdx.x;
    if (i >= total) return;
    const int node = i >> 6;   // HID == 64
    const int f    = i & 63;
    const float d  = dinv[node];
    const float v  = agg[i] + hlin[i] * d * d + bias[f];
    agg[i] = fmaxf(v, 0.0f);
}

// ---------------------------------------------------------------------------
// Global mean pool, stage 1: atomic sums per (graph, feature) + counts.
// ---------------------------------------------------------------------------
__global__ void pool_kernel(const float* __restrict__ h,
                            const long long* __restrict__ batch,
                            float* sums, float* counts, int total)
{
    int i = blockIdx.x * blockDim.x + threadIdx.x;
    if (i >= total) return;
    const int node = i >> 6;
    const int f    = i & 63;
    const int g    = (int)batch[node];
    unsafeAtomicAdd(&sums[g * HID + f], h[i]);
    if (f == 0) unsafeAtomicAdd(&counts[g], 1.0f);
}

// ---------------------------------------------------------------------------
// Head: pooled mean -> linear(64,2) -> log_softmax. One thread per graph.
// ---------------------------------------------------------------------------
__global__ void head_kernel(const float* __restrict__ sums,
                            const float* __restrict__ counts,
                            const float* __restrict__ Wl,
                            const float* __restrict__ bl,
                            float* out)
{
    int g = threadIdx.x;
    if (g >= N_GRAPHS) return;
    const float c = fmaxf(counts[g], 1.0f);
    float l0 = bl[0], l1 = bl[1];
#pragma unroll 8
    for (int f = 0; f < HID; ++f) {
        const float p = sums[g * HID + f] / c;
        l0 += p * Wl[f * 2 + 0];
        l1 += p * Wl[f * 2 + 1];
    }
    const float m   = fmaxf(l0, l1);
    const float lse = m + logf(expf(l0 - m) + expf(l1 - m));
    out[g * 2 + 0] = l0 - lse;
    out[g * 2 + 1] = l1 - lse;
}

// ---------------------------------------------------------------------------

extern "C" void kernel_launch(void* const* d_in, const int* in_sizes, int n_in,
                              void* d_out, int out_size, void* d_ws, size_t ws_size,
                              hipStream_t stream)
{
    const float*     x     = (const float*)d_in[0];
    const long long* ei    = (const long long*)d_in[1];   // [2, E] int64
    const long long* batch = (const long long*)d_in[2];   // [N] int64
    const float*     W1    = (const float*)d_in[3];
    const float*     b1    = (const float*)d_in[4];
    const float*     W2    = (const float*)d_in[5];
    const float*     b2    = (const float*)d_in[6];
    const float*     Wl    = (const float*)d_in[7];
    const float*     bl    = (const float*)d_in[8];
    float*           out   = (float*)d_out;

    const long long* row = ei;            // sources
    const long long* col = ei + N_EDGES;  // targets

    // Workspace carve-up (floats)
    const size_t NH = (size_t)N_NODES * HID;
    float* bufA   = (float*)d_ws;          // GEMM outputs (h_lin)
    float* bufB   = bufA + NH;             // aggregation / activations
    float* dinv   = bufB + NH;             // [N] deg -> rsqrt(deg)
    float* pooled = dinv + N_NODES;        // [G*HID]
    float* counts = pooled + (size_t)N_GRAPHS * HID;  // [G]

    const int totNH = N_NODES * HID;      // 3.2M, fits int
    const int eThreads = N_EDGES * (HID / 4);

    // --- degrees / normalization ---
    deg_init_kernel<<<(N_NODES + 255) / 256, 256, 0, stream>>>(dinv, N_NODES);
    deg_edges_kernel<<<(N_EDGES + 255) / 256, 256, 0, stream>>>(col, dinv, N_EDGES);
    rsqrt_inplace_kernel<<<(N_NODES + 255) / 256, 256, 0, stream>>>(dinv, N_NODES);

    // --- layer 1: h_lin = x @ W1 ---
    gemm_wmma_f32<<<N_NODES / 16, 128, 0, stream>>>(x, W1, bufA, N_NODES, HID, IN_DIM);
    hipMemsetAsync(bufB, 0, NH * sizeof(float), stream);
    edge_scatter_kernel<<<(eThreads + 255) / 256, 256, 0, stream>>>(row, col, dinv, bufA, bufB, N_EDGES);
    epilogue_kernel<<<(totNH + 255) / 256, 256, 0, stream>>>(bufB, bufA, dinv, b1, totNH);
    // bufB now holds h1 (post-ReLU)

    // --- layer 2: h_lin = h1 @ W2 ---
    gemm_wmma_f32<<<N_NODES / 16, 128, 0, stream>>>(bufB, W2, bufA, N_NODES, HID, HID);
    hipMemsetAsync(bufB, 0, NH * sizeof(float), stream);
    edge_scatter_kernel<<<(eThreads + 255) / 256, 256, 0, stream>>>(row, col, dinv, bufA, bufB, N_EDGES);
    epilogue_kernel<<<(totNH + 255) / 256, 256, 0, stream>>>(bufB, bufA, dinv, b2, totNH);
    // bufB now holds h2 (post-ReLU)

    // --- pooling + head ---
    hipMemsetAsync(pooled, 0, (size_t)(N_GRAPHS * HID + N_GRAPHS) * sizeof(float), stream);
    pool_kernel<<<(totNH + 255) / 256, 256, 0, stream>>>(bufB, batch, pooled, counts, totNH);
    head_kernel<<<1, 64, 0, stream>>>(pooled, counts, Wl, bl, out);
}